// SphericalHarmonicEdgeAttrs_42193758716383
// MI455X (gfx1250) — compile-verified
//
#include <hip/hip_runtime.h>
#include <math.h>

typedef __attribute__((ext_vector_type(2))) float v2f;
typedef __attribute__((ext_vector_type(4))) float f4;
typedef __attribute__((ext_vector_type(8))) float v8f;

#if defined(__has_builtin)
#if __has_builtin(__builtin_amdgcn_wmma_f32_16x16x4_f32)
#define HAVE_WMMA_F32X4 1
#endif
#if __has_builtin(__builtin_amdgcn_rsqf)
#define HAVE_RSQ 1
#endif
#endif
#ifndef HAVE_WMMA_F32X4
#pragma message("wmma_f32_16x16x4_f32 builtin NOT available -- using FMA fallback")
#define HAVE_WMMA_F32X4 0
#endif
#ifndef HAVE_RSQ
#pragma message("rsqf builtin NOT available -- using 1/sqrtf fallback")
#define HAVE_RSQ 0
#endif

// ds_swizzle group-of-32: xor_mask=0x10, or_mask=0, and_mask=0x1f -> swap lane halves
#define SWZ_SWAPX16 0x401F

__device__ __forceinline__ float swap16(float v) {
    return __int_as_float(__builtin_amdgcn_ds_swizzle(__float_as_int(v), SWZ_SWAPX16));
}

__global__ void __launch_bounds__(256)
sh_edge_attrs_kernel(const float* __restrict__ pos,
                     const int*   __restrict__ edge_index,
                     const float* __restrict__ cell,
                     const float* __restrict__ shift,
                     float*       __restrict__ out,
                     int E) {
    const int gid = blockIdx.x * blockDim.x + threadIdx.x;
    // clamp instead of early-return: keeps EXEC all-ones for WMMA (E%256==0 in
    // practice; if it ever isn't, duplicated lanes write identical values ->
    // deterministic).
    const int e = (gid < E) ? gid : (E - 1);
    const int t = threadIdx.x & 31;

    // ---- streaming loads (non-temporal: consumed once) ----
    const int src = __builtin_nontemporal_load(edge_index + e);
    const int dst = __builtin_nontemporal_load(edge_index + E + e);
    const size_t e3 = (size_t)e * 3;
    const float sx = __builtin_nontemporal_load(shift + e3 + 0);
    const float sy = __builtin_nontemporal_load(shift + e3 + 1);
    const float sz = __builtin_nontemporal_load(shift + e3 + 2);

    // ---- cached gather from pos (1.2 MB -> lives in L2/WGP$) ----
    const size_t s3 = (size_t)src * 3, d3 = (size_t)dst * 3;
    const float dx = pos[d3 + 0] - pos[s3 + 0];
    const float dy = pos[d3 + 1] - pos[s3 + 1];
    const float dz = pos[d3 + 2] - pos[s3 + 2];

    // ---- shift @ cell  (einsum 'ni,ij->nj') ----
    float cxv, cyv, czv;
#if HAVE_WMMA_F32X4
    // D = A x B with A = cell^T padded to 16x4, B = shift^T padded 4x16, so
    // D[m,n] = sum_k cell[k][m] * shift[edge n][k] = (shift@cell) component m.
    // A layout (16x4 f32): lane m: V0=A[m,0], V1=A[m,1]; lane m+16: V0=A[m,2], V1=0
    float a0 = 0.0f, a1 = 0.0f;
    {
        const int m = t & 15;
        if (m < 3) {
            a0 = cell[((t < 16) ? 0 : 6) + m];          // cell[0][m] or cell[2][m]
            a1 = (t < 16) ? cell[3 + m] : 0.0f;         // cell[1][m]
        }
    }
    v2f A; A.x = a0; A.y = a1;

    const bool lo = (t < 16);
    const float szw = swap16(sz);
    const float sxw = swap16(sx);
    const float syw = swap16(sy);
    // B layout (4x16 f32): V0 = {K=0 | K=2}, V1 = {K=1 | K=3} across lane halves
    v2f B1; B1.x = lo ? sx  : szw; B1.y = lo ? sy  : 0.0f;  // edges (base+0..15)
    v2f B2; B2.x = lo ? sxw : sz;  B2.y = lo ? syw : 0.0f;  // edges (base+16..31)

    v8f Z = {};
    v8f D1 = __builtin_amdgcn_wmma_f32_16x16x4_f32(false, A, false, B1,
                                                   (short)0, Z, false, false);
    v8f D2 = __builtin_amdgcn_wmma_f32_16x16x4_f32(false, A, false, B2,
                                                   (short)0, Z, false, false);
    // D layout: VGPR r holds M=r across lanes 0-15 (N=lane). Rows m<3 = result.
    const float w0 = swap16(D2[0]);
    const float w1 = swap16(D2[1]);
    const float w2 = swap16(D2[2]);
    cxv = lo ? D1[0] : w0;
    cyv = lo ? D1[1] : w1;
    czv = lo ? D1[2] : w2;
#else
    cxv = fmaf(sx, cell[0], fmaf(sy, cell[3], sz * cell[6]));
    cyv = fmaf(sx, cell[1], fmaf(sy, cell[4], sz * cell[7]));
    czv = fmaf(sx, cell[2], fmaf(sy, cell[5], sz * cell[8]));
#endif

    const float vx = dx + cxv;
    const float vy = dy + cyv;
    const float vz = dz + czv;

    // ---- normalize: u = v / max(|v|, 1e-12)  ==  v * rsq(max(|v|^2, 1e-24))
    // (1e-24f is a normal f32; max commutes with sqrt). Single v_rsq_f32
    // instead of ~30-op IEEE sqrt+div expansion.
    const float r2 = fmaf(vx, vx, fmaf(vy, vy, vz * vz));
#if HAVE_RSQ
    const float inv = __builtin_amdgcn_rsqf(fmaxf(r2, 1e-24f));
#else
    const float inv = 1.0f / sqrtf(fmaxf(r2, 1e-24f));
#endif
    const float x = vx * inv, y = vy * inv, z = vz * inv;

    // ---- spherical harmonics, lmax=3, component norm ----
    const float SQRT3 = 1.7320508075688772f;
    const float SQRT5 = 2.2360679774997896f;
    const float SQRT7 = 2.6457513110645906f;
    const float A3 = 1.0801234497346435f;   // sqrt(42)/6
    const float B3 = 1.6201851746019651f;   // sqrt(168)/8

    const float x2 = x * x, y2 = y * y, z2 = z * z;
    const float x2z2 = x2 + z2;

    const float sh2_0 = SQRT3 * x * z;
    const float sh2_1 = SQRT3 * x * y;
    const float sh2_2 = y2 - 0.5f * x2z2;
    const float sh2_3 = SQRT3 * y * z;
    const float sh2_4 = 0.5f * SQRT3 * (z2 - x2);

    const float t4 = 4.0f * y2 - x2z2;
    const float sh3_0 = A3 * (sh2_0 * z + sh2_4 * x);
    const float sh3_1 = SQRT7 * sh2_0 * y;
    const float sh3_2 = B3 * t4 * x;
    const float sh3_3 = 0.5f * SQRT7 * y * (2.0f * y2 - 3.0f * x2z2);
    const float sh3_4 = B3 * z * t4;
    const float sh3_5 = SQRT7 * sh2_4 * y;
    const float sh3_6 = A3 * (sh2_4 * z - sh2_0 * x);

    // ---- 4 x 128-bit non-temporal stores (64 B/edge, 64 B aligned) ----
    float* op = out + (size_t)e * 16;
    f4 o0 = {1.0f,            SQRT3 * x,       SQRT3 * y,       SQRT3 * z};
    f4 o1 = {SQRT5 * sh2_0,   SQRT5 * sh2_1,   SQRT5 * sh2_2,   SQRT5 * sh2_3};
    f4 o2 = {SQRT5 * sh2_4,   SQRT7 * sh3_0,   SQRT7 * sh3_1,   SQRT7 * sh3_2};
    f4 o3 = {SQRT7 * sh3_3,   SQRT7 * sh3_4,   SQRT7 * sh3_5,   SQRT7 * sh3_6};
    __builtin_nontemporal_store(o0, (f4*)(op + 0));
    __builtin_nontemporal_store(o1, (f4*)(op + 4));
    __builtin_nontemporal_store(o2, (f4*)(op + 8));
    __builtin_nontemporal_store(o3, (f4*)(op + 12));
}

extern "C" void kernel_launch(void* const* d_in, const int* in_sizes, int n_in,
                              void* d_out, int out_size, void* d_ws, size_t ws_size,
                              hipStream_t stream) {
    const float* pos        = (const float*)d_in[0];
    const int*   edge_index = (const int*)  d_in[1];  // [2, E]
    const float* cell       = (const float*)d_in[2];  // [1, 3, 3]
    const float* shift      = (const float*)d_in[3];  // [E, 3]
    float* out = (float*)d_out;                       // [E, 16]

    const int E = in_sizes[1] / 2;
    const int threads = 256;                          // 8 wave32s per block
    const int blocks = (E + threads - 1) / threads;
    sh_edge_attrs_kernel<<<blocks, threads, 0, stream>>>(pos, edge_index, cell,
                                                         shift, out, E);
}